// InvariantPointAttention_7919919694427
// MI455X (gfx1250) — compile-verified
//
#include <hip/hip_runtime.h>
#include <hip/hip_bf16.h>
#include <math.h>

#define NN    768
#define C_S   384
#define C_Zc  128
#define C_Hc  16
#define Hh    12
#define PQc   4
#define PVc   8
#define CAT_DIM 2112   // H*(C_Z + C_H + PV*4)
#define INF_V 100000.0f
#define EPS_V 1e-8f

typedef __attribute__((ext_vector_type(2))) float v2f;
typedef __attribute__((ext_vector_type(8))) float v8f;

// ---- workspace layout (float offsets) ----
#define WS_Q     0
#define WS_K     (WS_Q    + Hh*NN*C_Hc)      // [H][N][16]
#define WS_VCAT  (WS_K    + Hh*NN*C_Hc)      // [H][N][48]: 0..15 v, 16..39 v_pts(p*3+x), 40..47 pad=0
#define WS_QPTS  (WS_VCAT + Hh*NN*48)        // [N][H][PQ][3]
#define WS_KPTS  (WS_QPTS + NN*Hh*PQc*3)     // [N][H][PQ][3]
#define WS_BB    (WS_KPTS + NN*Hh*PQc*3)     // [H][N][N]
#define WS_ATT   (WS_BB   + Hh*NN*NN)        // [H][N][N] logits -> softmax (in place)
#define WS_AOUT  (WS_ATT  + Hh*NN*NN)        // [H][N][48]
#define WS_CAT   (WS_AOUT + Hh*NN*48)        // [N][2112]

static __device__ inline v8f wmma_f32_k4(v2f a, v2f b, v8f c) {
  return __builtin_amdgcn_wmma_f32_16x16x4_f32(false, a, false, b, (short)0, c,
                                               false, false);
}

// ================= K0: projections =================
__global__ void ipa_proj(const float* __restrict__ s, const float* __restrict__ rot,
                         const float* __restrict__ trans,
                         const float* __restrict__ Wq,  const float* __restrict__ bq,
                         const float* __restrict__ Wkv, const float* __restrict__ bkv,
                         const float* __restrict__ Wqp, const float* __restrict__ bqp,
                         const float* __restrict__ Wkvp,const float* __restrict__ bkvp,
                         float* __restrict__ ws) {
  __shared__ float sS[C_S];
  const int i = blockIdx.x;
  const int t = threadIdx.x;
  for (int c = t; c < C_S; c += 256) sS[c] = s[i * C_S + c];
  __syncthreads();

  float* q  = ws + WS_Q;
  float* k  = ws + WS_K;
  float* vc = ws + WS_VCAT;
  float* qp = ws + WS_QPTS;
  float* kp = ws + WS_KPTS;

  for (int task = t; task < 768; task += 256) {
    if (task < 192) {                       // q
      int c = task;
      float acc = bq[c];
      const float* w = Wq + (size_t)c * C_S;
      for (int u = 0; u < C_S; ++u) acc += sS[u] * w[u];
      int h = c >> 4, ch = c & 15;
      q[((size_t)h * NN + i) * C_Hc + ch] = acc;
    } else if (task < 576) {                // kv
      int r = task - 192;
      float acc = bkv[r];
      const float* w = Wkv + (size_t)r * C_S;
      for (int u = 0; u < C_S; ++u) acc += sS[u] * w[u];
      int h = r >> 5, cc = r & 31;
      if (cc < 16) k[((size_t)h * NN + i) * C_Hc + cc] = acc;
      else         vc[((size_t)h * NN + i) * 48 + (cc - 16)] = acc;
    } else if (task < 624) {                // q_pts (48 points)
      int d = task - 576;
      float p3[3];
      for (int x = 0; x < 3; ++x) {
        int row = x * 48 + d;
        float acc = bqp[row];
        const float* w = Wqp + (size_t)row * C_S;
        for (int u = 0; u < C_S; ++u) acc += sS[u] * w[u];
        p3[x] = acc;
      }
      const float* R = rot + (size_t)i * 9;
      float gx = R[0]*p3[0] + R[1]*p3[1] + R[2]*p3[2] + trans[i*3+0];
      float gy = R[3]*p3[0] + R[4]*p3[1] + R[5]*p3[2] + trans[i*3+1];
      float gz = R[6]*p3[0] + R[7]*p3[1] + R[8]*p3[2] + trans[i*3+2];
      int h = d / PQc, pp = d % PQc;
      float* dst = qp + (((size_t)i * Hh + h) * PQc + pp) * 3;
      dst[0] = gx; dst[1] = gy; dst[2] = gz;
    } else {                                // kv_pts (144 points)
      int d = task - 624;
      float p3[3];
      for (int x = 0; x < 3; ++x) {
        int row = x * 144 + d;
        float acc = bkvp[row];
        const float* w = Wkvp + (size_t)row * C_S;
        for (int u = 0; u < C_S; ++u) acc += sS[u] * w[u];
        p3[x] = acc;
      }
      const float* R = rot + (size_t)i * 9;
      float g[3];
      g[0] = R[0]*p3[0] + R[1]*p3[1] + R[2]*p3[2] + trans[i*3+0];
      g[1] = R[3]*p3[0] + R[4]*p3[1] + R[5]*p3[2] + trans[i*3+1];
      g[2] = R[6]*p3[0] + R[7]*p3[1] + R[8]*p3[2] + trans[i*3+2];
      int h = d / 12, pt = d % 12;
      if (pt < PQc) {
        float* dst = kp + (((size_t)i * Hh + h) * PQc + pt) * 3;
        dst[0] = g[0]; dst[1] = g[1]; dst[2] = g[2];
      } else {
        float* dst = vc + ((size_t)h * NN + i) * 48 + 16 + (pt - PQc) * 3;
        dst[0] = g[0]; dst[1] = g[1]; dst[2] = g[2];
      }
    }
  }
  if (t < Hh * 8) {                          // zero the 8 pad cols of vcat
    int h = t >> 3, c = 40 + (t & 7);
    vc[((size_t)h * NN + i) * 48 + c] = 0.0f;
  }
}

// ================= K1: pair bias (z pass 1) =================
__global__ void ipa_bias(const float* __restrict__ z, const float* __restrict__ Wb,
                         const float* __restrict__ bb, float* __restrict__ ws) {
  __shared__ float sW[Hh * C_Zc];
  const int i = blockIdx.x;
  const int j = blockIdx.y * 256 + threadIdx.x;
  for (int c = threadIdx.x; c < Hh * C_Zc; c += 256) sW[c] = Wb[c];
  __syncthreads();
  const float4* zr = (const float4*)(z + ((size_t)i * NN + j) * C_Zc);
  float acc[Hh];
  for (int h = 0; h < Hh; ++h) acc[h] = bb[h];
  for (int c4 = 0; c4 < C_Zc / 4; ++c4) {
    float4 zv = zr[c4];
    #pragma unroll
    for (int h = 0; h < Hh; ++h) {
      const float* w = sW + h * C_Zc + c4 * 4;
      acc[h] += zv.x * w[0] + zv.y * w[1] + zv.z * w[2] + zv.w * w[3];
    }
  }
  float* bbw = ws + WS_BB;
  for (int h = 0; h < Hh; ++h)
    bbw[((size_t)h * NN + i) * NN + j] = acc[h];
}

// ================= K2: attention logits (WMMA q.kT) =================
__global__ void ipa_logits(const float* __restrict__ mask,
                           const float* __restrict__ hwts,
                           float* __restrict__ ws) {
  const int h  = blockIdx.x;
  const int it = blockIdx.y * 16;
  const int jt = blockIdx.z * 16;
  const int lane = threadIdx.x;
  const int ml = lane & 15, kh = lane >> 4;

  const float* q = ws + WS_Q + (size_t)h * NN * C_Hc;
  const float* k = ws + WS_K + (size_t)h * NN * C_Hc;

  v8f acc = {};
  #pragma unroll
  for (int kb = 0; kb < 4; ++kb) {
    int kk = kb * 4 + 2 * kh;
    v2f A, B;
    A.x = q[(size_t)(it + ml) * C_Hc + kk];
    A.y = q[(size_t)(it + ml) * C_Hc + kk + 1];
    B.x = k[(size_t)(jt + ml) * C_Hc + kk];
    B.y = k[(size_t)(jt + ml) * C_Hc + kk + 1];
    acc = wmma_f32_k4(A, B, acc);
  }

  const float hwv = log1pf(expf(hwts[h])) * sqrtf(1.0f / 54.0f); // 3*(PQ*9/2)=54
  const float sc1 = 0.14433756729740643f;  // sqrt(1/48)
  const float scb = 0.5773502691896258f;   // sqrt(1/3)
  const int j = jt + ml;
  float kpx[PQc * 3];
  {
    const float* kpb = ws + WS_KPTS + ((size_t)j * Hh + h) * PQc * 3;
    #pragma unroll
    for (int u = 0; u < PQc * 3; ++u) kpx[u] = kpb[u];
  }
  const float mj = mask[j];
  float* att       = ws + WS_ATT + (size_t)h * NN * NN;
  const float* bbw = ws + WS_BB  + (size_t)h * NN * NN;

  #pragma unroll
  for (int r = 0; r < 8; ++r) {
    int i = it + r + 8 * kh;
    const float* qpb = ws + WS_QPTS + ((size_t)i * Hh + h) * PQc * 3;
    float pt = 0.0f;
    #pragma unroll
    for (int p = 0; p < PQc; ++p) {
      float dx = qpb[p*3+0] - kpx[p*3+0];
      float dy = qpb[p*3+1] - kpx[p*3+1];
      float dz = qpb[p*3+2] - kpx[p*3+2];
      pt += dx*dx + dy*dy + dz*dz;
    }
    float val = acc[r] * sc1 + scb * bbw[(size_t)i * NN + j]
              - 0.5f * hwv * pt + INF_V * (mask[i] * mj - 1.0f);
    att[(size_t)i * NN + j] = val;
  }
}

// ================= K3: row softmax =================
__global__ void ipa_softmax(float* __restrict__ ws) {
  __shared__ float red[256];
  const int row = blockIdx.x;                // h*NN + i
  float* a = ws + WS_ATT + (size_t)row * NN;
  const int t = threadIdx.x;

  float m = -3.4e38f;
  for (int j = t; j < NN; j += 256) m = fmaxf(m, a[j]);
  red[t] = m; __syncthreads();
  for (int s = 128; s > 0; s >>= 1) {
    if (t < s) red[t] = fmaxf(red[t], red[t + s]);
    __syncthreads();
  }
  m = red[0]; __syncthreads();

  float sum = 0.0f;
  for (int j = t; j < NN; j += 256) {
    float e = expf(a[j] - m);
    a[j] = e;
    sum += e;
  }
  red[t] = sum; __syncthreads();
  for (int s = 128; s > 0; s >>= 1) {
    if (t < s) red[t] += red[t + s];
    __syncthreads();
  }
  const float inv = 1.0f / red[0];
  for (int j = t; j < NN; j += 256) a[j] *= inv;
}

// ================= K4a: o and o_pt sums (WMMA a@[v|v_pts]) =================
__global__ void ipa_av(float* __restrict__ ws) {
  const int h  = blockIdx.x;
  const int it = blockIdx.y * 16;
  const int ct = blockIdx.z * 16;
  const int lane = threadIdx.x;
  const int ml = lane & 15, kh = lane >> 4;

  const float* a  = ws + WS_ATT  + (size_t)h * NN * NN;
  const float* vc = ws + WS_VCAT + (size_t)h * NN * 48;

  v8f acc = {};
  #pragma unroll 1
  for (int kb = 0; kb < NN / 4; ++kb) {
    int kk = kb * 4 + 2 * kh;
    v2f A, B;
    A.x = a[(size_t)(it + ml) * NN + kk];
    A.y = a[(size_t)(it + ml) * NN + kk + 1];
    B.x = vc[(size_t)kk       * 48 + ct + ml];
    B.y = vc[(size_t)(kk + 1) * 48 + ct + ml];
    acc = wmma_f32_k4(A, B, acc);
  }
  float* ao = ws + WS_AOUT + (size_t)h * NN * 48;
  #pragma unroll
  for (int r = 0; r < 8; ++r)
    ao[(size_t)(it + r + 8 * kh) * 48 + ct + ml] = acc[r];
}

// ================= K4b: o_pair (z pass 2, WMMA per residue) =================
__global__ void ipa_opair(const float* __restrict__ z, float* __restrict__ ws) {
  const int i = blockIdx.x;
  const int lane = threadIdx.x;
  const int ml = lane & 15, kh = lane >> 4;
  const float* a  = ws + WS_ATT;             // [H][N][N]
  const float* zr = z + (size_t)i * NN * C_Zc;

  v8f zero = {};
  v8f acc[8];
  #pragma unroll
  for (int nt = 0; nt < 8; ++nt) acc[nt] = zero;

  #pragma unroll 1
  for (int kb = 0; kb < NN / 4; ++kb) {
    int kk = kb * 4 + 2 * kh;
    v2f A;
    if (ml < Hh) {
      const float* ar = a + ((size_t)ml * NN + i) * NN;
      A.x = ar[kk]; A.y = ar[kk + 1];
    } else {
      A.x = 0.0f; A.y = 0.0f;
    }
    #pragma unroll
    for (int nt = 0; nt < 8; ++nt) {
      v2f B;
      B.x = zr[(size_t)kk       * C_Zc + nt * 16 + ml];
      B.y = zr[(size_t)(kk + 1) * C_Zc + nt * 16 + ml];
      acc[nt] = wmma_f32_k4(A, B, acc[nt]);
    }
  }

  float* cat = ws + WS_CAT + (size_t)i * CAT_DIM;
  #pragma unroll
  for (int nt = 0; nt < 8; ++nt) {
    #pragma unroll
    for (int r = 0; r < 8; ++r) {
      int m = r + 8 * kh;                    // head index
      if (m < Hh) cat[576 + m * C_Zc + nt * 16 + ml] = acc[nt][r];
    }
  }
}

// ================= K4c: inverse transform, norms, assemble cat =================
__global__ void ipa_post(const float* __restrict__ rot, const float* __restrict__ trans,
                         float* __restrict__ ws) {
  const int i = blockIdx.x;
  const int t = threadIdx.x;
  const float* ao = ws + WS_AOUT;
  float* cat = ws + WS_CAT + (size_t)i * CAT_DIM;

  for (int idx = t; idx < Hh * C_Hc; idx += 128) {        // o
    int h = idx >> 4, c = idx & 15;
    cat[idx] = ao[((size_t)h * NN + i) * 48 + c];
  }
  if (t < Hh * PVc) {                                     // o_pt + norm
    int h = t >> 3, p = t & 7;
    const float* b = ao + ((size_t)h * NN + i) * 48 + 16 + p * 3;
    float dx = b[0] - trans[i*3+0];
    float dy = b[1] - trans[i*3+1];
    float dz = b[2] - trans[i*3+2];
    const float* R = rot + (size_t)i * 9;
    float lx = R[0]*dx + R[3]*dy + R[6]*dz;               // R^T . d
    float ly = R[1]*dx + R[4]*dy + R[7]*dz;
    float lz = R[2]*dx + R[5]*dy + R[8]*dz;
    float nrm = sqrtf(lx*lx + ly*ly + lz*lz + EPS_V);
    cat[192 + t] = lx;
    cat[288 + t] = ly;
    cat[384 + t] = lz;
    cat[480 + t] = nrm;
  }
}

// ================= K5: final GEMM cat @ Wout^T + bout =================
__global__ void ipa_out(const float* __restrict__ Wout, const float* __restrict__ bout,
                        const float* __restrict__ ws, float* __restrict__ out) {
  const int it = blockIdx.x * 16;
  const int mt = blockIdx.y * 16;
  const int lane = threadIdx.x;
  const int ml = lane & 15, kh = lane >> 4;
  const float* cat = ws + WS_CAT;

  v8f acc = {};
  #pragma unroll 1
  for (int kb = 0; kb < CAT_DIM / 4; ++kb) {
    int kk = kb * 4 + 2 * kh;
    v2f A, B;
    A.x = cat[(size_t)(it + ml) * CAT_DIM + kk];
    A.y = cat[(size_t)(it + ml) * CAT_DIM + kk + 1];
    B.x = Wout[(size_t)(mt + ml) * CAT_DIM + kk];
    B.y = Wout[(size_t)(mt + ml) * CAT_DIM + kk + 1];
    acc = wmma_f32_k4(A, B, acc);
  }
  #pragma unroll
  for (int r = 0; r < 8; ++r) {
    int i = it + r + 8 * kh;
    out[(size_t)i * C_S + mt + ml] = acc[r] + bout[mt + ml];
  }
}

extern "C" void kernel_launch(void* const* d_in, const int* in_sizes, int n_in,
                              void* d_out, int out_size, void* d_ws, size_t ws_size,
                              hipStream_t stream) {
  (void)in_sizes; (void)n_in; (void)out_size; (void)ws_size;
  const float* s     = (const float*)d_in[0];
  const float* z     = (const float*)d_in[1];
  const float* rot   = (const float*)d_in[2];
  const float* trans = (const float*)d_in[3];
  const float* mask  = (const float*)d_in[4];
  const float* Wq    = (const float*)d_in[5];
  const float* bq    = (const float*)d_in[6];
  const float* Wkv   = (const float*)d_in[7];
  const float* bkv   = (const float*)d_in[8];
  const float* Wqp   = (const float*)d_in[9];
  const float* bqp   = (const float*)d_in[10];
  const float* Wkvp  = (const float*)d_in[11];
  const float* bkvp  = (const float*)d_in[12];
  const float* Wb    = (const float*)d_in[13];
  const float* bb    = (const float*)d_in[14];
  const float* hw    = (const float*)d_in[15];
  const float* Wout  = (const float*)d_in[16];
  const float* bout  = (const float*)d_in[17];
  float* ws  = (float*)d_ws;
  float* out = (float*)d_out;

  ipa_proj   <<<NN, 256, 0, stream>>>(s, rot, trans, Wq, bq, Wkv, bkv,
                                      Wqp, bqp, Wkvp, bkvp, ws);
  ipa_bias   <<<dim3(NN, NN / 256), 256, 0, stream>>>(z, Wb, bb, ws);
  ipa_logits <<<dim3(Hh, NN / 16, NN / 16), 32, 0, stream>>>(mask, hw, ws);
  ipa_softmax<<<Hh * NN, 256, 0, stream>>>(ws);
  ipa_av     <<<dim3(Hh, NN / 16, 3), 32, 0, stream>>>(ws);
  ipa_opair  <<<NN, 32, 0, stream>>>(z, ws);
  ipa_post   <<<NN, 128, 0, stream>>>(rot, trans, ws);
  ipa_out    <<<dim3(NN / 16, C_S / 16), 32, 0, stream>>>(Wout, bout, ws, out);
}